// GraphSageEncoder_66975720014389
// MI455X (gfx1250) — compile-verified
//
#include <hip/hip_runtime.h>
#include <hip/hip_bf16.h>

typedef __attribute__((ext_vector_type(2))) float v2f;
typedef __attribute__((ext_vector_type(8))) float v8f;

// Builtins are only declared for the device (amdgcn) compilation pass;
// checking on the host pass always fails. Only assert device-side.
#if defined(__HIP_DEVICE_COMPILE__)
#if !__has_builtin(__builtin_amdgcn_wmma_f32_16x16x4_f32)
#error "wmma_f32_16x16x4_f32 builtin not available on device toolchain"
#endif
#endif

// ---------------- zero workspace region ----------------
__global__ void sage_zero_kernel(float* __restrict__ p, long long n) {
    long long i = (long long)blockIdx.x * blockDim.x + threadIdx.x;
    long long stride = (long long)gridDim.x * blockDim.x;
    for (; i < n; i += stride) p[i] = 0.0f;
}

// ---------------- per-node in-degree counts ----------------
__global__ void sage_count_kernel(const long long* __restrict__ ei, long long E,
                                  float* __restrict__ cnt) {
    long long i = (long long)blockIdx.x * blockDim.x + threadIdx.x;
    if (i >= E) return;
    long long d = ei[E + i];           // dst row of edge_index
    atomicAdd(&cnt[d], 1.0f);
}

// ---------------- scatter-add messages: S[dst] += X[src] ----------------
// One thread = one edge x 4 channels (float4 gather, 4 f32 atomics).
__global__ void sage_scatter_kernel(const long long* __restrict__ ei, long long E,
                                    const float* __restrict__ X,
                                    float* __restrict__ S) {
    const int C = 64;
    long long idx = (long long)blockIdx.x * blockDim.x + threadIdx.x;
    long long total = E * (C / 4);
    if (idx >= total) return;
    long long e = idx >> 4;            // /16
    int q = (int)(idx & 15);
    long long s = ei[e];
    long long d = ei[E + e];
    const float4 v = ((const float4*)(X + s * C))[q];
    float* o = S + d * C + (long long)q * 4;
    atomicAdd(o + 0, v.x);
    atomicAdd(o + 1, v.y);
    atomicAdd(o + 2, v.z);
    atomicAdd(o + 3, v.w);
}

// ---------------- fused SAGE layer: out = [relu]( (S/cnt)@Wl + X@Wr + b ) ----
// One wave per 16-node tile; f32 WMMA 16x16x4; weights staged in LDS.
template <int CIN, int COUT, bool RELU>
__global__ void sage_gemm_kernel(const float* __restrict__ X,
                                 const float* __restrict__ S,
                                 const float* __restrict__ cnt,
                                 const float* __restrict__ Wl,
                                 const float* __restrict__ Wr,
                                 const float* __restrict__ bias,
                                 float* __restrict__ out,
                                 int N, int ntiles) {
    __shared__ float sWl[CIN * COUT];
    __shared__ float sWr[CIN * COUT];
    __shared__ float sB[COUT];
    for (int i = threadIdx.x; i < CIN * COUT; i += blockDim.x) {
        sWl[i] = Wl[i];
        sWr[i] = Wr[i];
    }
    for (int i = threadIdx.x; i < COUT; i += blockDim.x) sB[i] = bias[i];
    __syncthreads();

    const int lane  = threadIdx.x & 31;
    const int wave  = threadIdx.x >> 5;
    const int tile  = blockIdx.x * (blockDim.x >> 5) + wave;
    if (tile >= ntiles) return;        // wave-uniform: EXEC stays all-ones

    const int row   = lane & 15;       // A row / C column index
    const int khalf = lane >> 4;       // which K-pair half this lane holds
    int node = tile * 16 + row;
    if (node >= N) node = N - 1;       // clamp loads for ragged tail
    const float invc = 1.0f / fmaxf(cnt[node], 1.0f);

    // Preload all A K-slices once (reused for every N tile).
    // A 16x4 f32 layout: lane holds A[row][4*kk + 2*khalf + j] in vgpr j.
    v2f aM[CIN / 4];  // mean-aggregated neighbors
    v2f aX[CIN / 4];  // self features
    const float* __restrict__ xrow = X + (long long)node * CIN;
    const float* __restrict__ srow = S + (long long)node * CIN;
#pragma unroll
    for (int kk = 0; kk < CIN / 4; ++kk) {
        const int k0 = kk * 4 + khalf * 2;
        aM[kk].x = srow[k0] * invc;
        aM[kk].y = srow[k0 + 1] * invc;
        aX[kk].x = xrow[k0];
        aX[kk].y = xrow[k0 + 1];
    }

#pragma unroll
    for (int nt = 0; nt < COUT / 16; ++nt) {
        const int col = nt * 16 + row;       // C/D column for this lane
        const float bv = sB[col];
        v8f c = {bv, bv, bv, bv, bv, bv, bv, bv};
#if defined(__HIP_DEVICE_COMPILE__)
        // term 1: mean-agg @ Wl
#pragma unroll
        for (int kk = 0; kk < CIN / 4; ++kk) {
            const int k0 = kk * 4 + khalf * 2;
            v2f b;
            b.x = sWl[k0 * COUT + col];
            b.y = sWl[(k0 + 1) * COUT + col];
            c = __builtin_amdgcn_wmma_f32_16x16x4_f32(
                    false, aM[kk], false, b, (short)0, c, false, false);
        }
        // term 2: x @ Wr
#pragma unroll
        for (int kk = 0; kk < CIN / 4; ++kk) {
            const int k0 = kk * 4 + khalf * 2;
            v2f b;
            b.x = sWr[k0 * COUT + col];
            b.y = sWr[(k0 + 1) * COUT + col];
            c = __builtin_amdgcn_wmma_f32_16x16x4_f32(
                    false, aX[kk], false, b, (short)0, c, false, false);
        }
#endif
        // C/D layout: vgpr r -> matrix row r + 8*khalf, column = lane&15
#pragma unroll
        for (int r = 0; r < 8; ++r) {
            float v = c[r];
            if (RELU) v = fmaxf(v, 0.0f);
            const int m = tile * 16 + r + khalf * 8;
            if (m < N) out[(long long)m * COUT + col] = v;
        }
    }
}

extern "C" void kernel_launch(void* const* d_in, const int* in_sizes, int n_in,
                              void* d_out, int out_size, void* d_ws, size_t ws_size,
                              hipStream_t stream) {
    const int CIN = 64, HID = 64, OUT = 32;
    const float*     x   = (const float*)d_in[0];
    const long long* ei  = (const long long*)d_in[1];
    const float*     Wl1 = (const float*)d_in[2];
    const float*     bl1 = (const float*)d_in[3];
    const float*     Wr1 = (const float*)d_in[4];
    const float*     Wl2 = (const float*)d_in[5];
    const float*     bl2 = (const float*)d_in[6];
    const float*     Wr2 = (const float*)d_in[7];
    float* out = (float*)d_out;

    const long long N = in_sizes[0] / CIN;        // 100000
    const long long E = in_sizes[1] / 2;          // 1600000

    // Workspace layout (floats): [cnt: N][S: N*64][h: N*64]
    float* cnt = (float*)d_ws;
    float* S   = cnt + N;
    float* h   = S + N * (long long)CIN;

    const int B = 256;
    const int ntiles = (int)((N + 15) / 16);
    const int waves_per_block = B / 32;
    const int gemm_blocks = (ntiles + waves_per_block - 1) / waves_per_block;

    // 1) zero cnt + S (contiguous), compute degrees
    {
        long long n = N + N * (long long)CIN;
        int blocks = (int)((n + B - 1) / B);
        if (blocks > 65535) blocks = 65535;
        sage_zero_kernel<<<blocks, B, 0, stream>>>(cnt, n);
    }
    sage_count_kernel<<<(int)((E + B - 1) / B), B, 0, stream>>>(ei, E, cnt);

    // 2) layer 1: scatter x, fused mean+GEMM+bias+relu -> h
    {
        long long total = E * (CIN / 4);
        sage_scatter_kernel<<<(int)((total + B - 1) / B), B, 0, stream>>>(ei, E, x, S);
    }
    sage_gemm_kernel<CIN, HID, true><<<gemm_blocks, B, 0, stream>>>(
        x, S, cnt, Wl1, Wr1, bl1, h, (int)N, ntiles);

    // 3) layer 2: re-zero S, scatter h, fused GEMM -> out
    {
        long long n = N * (long long)CIN;
        int blocks = (int)((n + B - 1) / B);
        if (blocks > 65535) blocks = 65535;
        sage_zero_kernel<<<blocks, B, 0, stream>>>(S, n);
    }
    {
        long long total = E * (CIN / 4);
        sage_scatter_kernel<<<(int)((total + B - 1) / B), B, 0, stream>>>(ei, E, h, S);
    }
    sage_gemm_kernel<HID, OUT, false><<<gemm_blocks, B, 0, stream>>>(
        h, S, cnt, Wl2, Wr2, bl2, out, (int)N, ntiles);
}